// GATModelPlaceholder_13340168421673
// MI455X (gfx1250) — compile-verified
//
#include <hip/hip_runtime.h>
#include <math.h>

// ---------------------------------------------------------------------------
// GAT 2-layer on MI455X (gfx1250, wave32).
// GEMMs via V_WMMA_F32_16X16X4_F32 (exact f32, matches reference precision).
// Edge aggregation via wave-per-edge float4 gathers + global_atomic_add_f32
// into L2-resident accumulators (T1/ACC1 = 51.2MB each, fit in 192MB L2).
// ---------------------------------------------------------------------------

typedef __attribute__((ext_vector_type(2))) float v2f;
typedef __attribute__((ext_vector_type(8))) float v8f;

#define N_NODES 50000
#define N_EDGES 1600000
#define D_IN    512
#define D_HEAD  64
#define D_H1    256   // 4 heads * 64

// -------------------------------------------------------------- utility ----
__global__ void zero_kernel(float* __restrict__ p, int n) {
  int i = blockIdx.x * blockDim.x + threadIdx.x;
  int stride = gridDim.x * blockDim.x;
  for (; i < n; i += stride) p[i] = 0.0f;
}

__global__ void count_kernel(const int* __restrict__ src, float* __restrict__ counts) {
  int i = blockIdx.x * blockDim.x + threadIdx.x;
  if (i < N_EDGES) atomicAdd(&counts[src[i]], 1.0f);
}

// ---------------------------------------------------------------- GEMM 1 ---
// T1[50000,256] = X[50000,512] @ W1cat[512,256]
// W1 stored (head, 512, 64); column c maps to W1[c>>6][k][c&63].
// One wave -> 16 rows x all 256 cols; 16 C fragments (128 VGPRs) live in regs.
__global__ __launch_bounds__(256) void gemm1_kernel(const float* __restrict__ X,
                                                    const float* __restrict__ W1,
                                                    float* __restrict__ T) {
  const int wave = blockIdx.x * (blockDim.x >> 5) + (threadIdx.x >> 5);
  if (wave >= N_NODES / 16) return;               // uniform per-wave branch
  const int lane = threadIdx.x & 31;
  const int half = lane >> 4;                     // 0: K=k0..k0+1, 1: K=k0+2..k0+3
  const int lidx = lane & 15;
  const int m0   = wave << 4;

  v8f zero = {};
  v8f acc[16];
#pragma unroll
  for (int i = 0; i < 16; ++i) acc[i] = zero;

  const float* xrow = X + (size_t)(m0 + lidx) * D_IN;
  for (int k0 = 0; k0 < D_IN; k0 += 4) {
    const int krow = k0 + 2 * half;
    v2f a = *(const v2f*)(xrow + krow);           // A frag: 2 f32 per lane
#pragma unroll
    for (int nt = 0; nt < 16; ++nt) {
      const float* wb = W1 + (size_t)(nt >> 2) * (D_IN * D_HEAD) + (nt & 3) * 16 + lidx;
      v2f b;
      b.x = wb[(size_t)(krow    ) * D_HEAD];
      b.y = wb[(size_t)(krow + 1) * D_HEAD];
      acc[nt] = __builtin_amdgcn_wmma_f32_16x16x4_f32(
          false, a, false, b, (short)0, acc[nt], false, false);
    }
  }
  // C layout: VGPR r -> lanes 0-15: M=m0+r, N=lidx; lanes 16-31: M=m0+8+r.
#pragma unroll
  for (int nt = 0; nt < 16; ++nt) {
    float* trow = T + (size_t)(m0 + half * 8) * D_H1 + nt * 16 + lidx;
#pragma unroll
    for (int r = 0; r < 8; ++r)
      trow[(size_t)r * D_H1] = acc[nt][r];
  }
}

// ---------------------------------------------------------------- GEMM 2 ---
// T2[50000,64] = H[50000,256] @ W2[256,64]
__global__ __launch_bounds__(256) void gemm2_kernel(const float* __restrict__ H,
                                                    const float* __restrict__ W2,
                                                    float* __restrict__ T2) {
  const int wave = blockIdx.x * (blockDim.x >> 5) + (threadIdx.x >> 5);
  if (wave >= N_NODES / 16) return;
  const int lane = threadIdx.x & 31;
  const int half = lane >> 4;
  const int lidx = lane & 15;
  const int m0   = wave << 4;

  v8f zero = {};
  v8f acc[4];
#pragma unroll
  for (int i = 0; i < 4; ++i) acc[i] = zero;

  const float* hrow = H + (size_t)(m0 + lidx) * D_H1;
  for (int k0 = 0; k0 < D_H1; k0 += 4) {
    const int krow = k0 + 2 * half;
    v2f a = *(const v2f*)(hrow + krow);
#pragma unroll
    for (int nt = 0; nt < 4; ++nt) {
      v2f b;
      b.x = W2[(size_t)(krow    ) * D_HEAD + nt * 16 + lidx];
      b.y = W2[(size_t)(krow + 1) * D_HEAD + nt * 16 + lidx];
      acc[nt] = __builtin_amdgcn_wmma_f32_16x16x4_f32(
          false, a, false, b, (short)0, acc[nt], false, false);
    }
  }
#pragma unroll
  for (int nt = 0; nt < 4; ++nt) {
    float* trow = T2 + (size_t)(m0 + half * 8) * D_HEAD + nt * 16 + lidx;
#pragma unroll
    for (int r = 0; r < 8; ++r)
      trow[(size_t)r * D_HEAD] = acc[nt][r];
  }
}

// -------------------------------------------------------------- scatters ---
// Layer1: one wave per edge; lane covers 8 of 256 feats (2x float4 gather).
__global__ void scatter1_kernel(const int* __restrict__ src, const int* __restrict__ dst,
                                const float* __restrict__ T, float* __restrict__ acc) {
  const int wave = blockIdx.x * (blockDim.x >> 5) + (threadIdx.x >> 5);
  if (wave >= N_EDGES) return;
  const int lane = threadIdx.x & 31;
  const int s = src[wave];
  const int d = dst[wave];
  const float4* trow = (const float4*)(T + (size_t)d * D_H1) + lane * 2;
  float* arow = acc + (size_t)s * D_H1 + lane * 8;
  float4 v0 = trow[0];
  float4 v1 = trow[1];
  atomicAdd(arow + 0, v0.x); atomicAdd(arow + 1, v0.y);
  atomicAdd(arow + 2, v0.z); atomicAdd(arow + 3, v0.w);
  atomicAdd(arow + 4, v1.x); atomicAdd(arow + 5, v1.y);
  atomicAdd(arow + 6, v1.z); atomicAdd(arow + 7, v1.w);
}

// Layer2: one wave per edge; lane covers 2 of 64 feats.
__global__ void scatter2_kernel(const int* __restrict__ src, const int* __restrict__ dst,
                                const float* __restrict__ T2, float* __restrict__ out) {
  const int wave = blockIdx.x * (blockDim.x >> 5) + (threadIdx.x >> 5);
  if (wave >= N_EDGES) return;
  const int lane = threadIdx.x & 31;
  const int s = src[wave];
  const int d = dst[wave];
  const float2* trow = (const float2*)(T2 + (size_t)d * D_HEAD) + lane;
  float2 v = *trow;
  float* orow = out + (size_t)s * D_HEAD + lane * 2;
  atomicAdd(orow + 0, v.x);
  atomicAdd(orow + 1, v.y);
}

// ----------------------------------------------------- pointwise kernels ---
__global__ void divide_elu_kernel(const float* __restrict__ acc,
                                  const float* __restrict__ counts,
                                  float* __restrict__ H) {
  int i = blockIdx.x * blockDim.x + threadIdx.x;
  if (i >= N_NODES * D_H1) return;
  float c = counts[i >> 8];                      // i / 256
  float h = (c > 0.0f) ? acc[i] / c : 0.0f;
  H[i] = (h > 0.0f) ? h : (expf(h) - 1.0f);     // elu, alpha=1
}

__global__ void finalize_kernel(float* __restrict__ out, const float* __restrict__ counts) {
  int i = blockIdx.x * blockDim.x + threadIdx.x;
  if (i >= N_NODES * D_HEAD) return;
  float c = counts[i >> 6];                      // i / 64
  out[i] = (c > 0.0f) ? out[i] / c : 0.0f;
}

// --------------------------------------------------------------- launch ----
extern "C" void kernel_launch(void* const* d_in, const int* in_sizes, int n_in,
                              void* d_out, int out_size, void* d_ws, size_t ws_size,
                              hipStream_t stream) {
  const float* X  = (const float*)d_in[0];           // (50000, 512)
  const int*   ei = (const int*)d_in[1];             // (2, 1600000)
  const float* W1 = (const float*)d_in[2];           // (4, 512, 64)
  const float* W2 = (const float*)d_in[3];           // (256, 64)
  float* out = (float*)d_out;                        // (50000, 64)

  const int* src = ei;
  const int* dst = ei + N_EDGES;

  // Workspace layout (floats): counts | bufA (T1 then H) | bufB (ACC1 then T2)
  float* counts = (float*)d_ws;
  float* bufA   = counts + 50176;                            // 12.8M floats
  float* bufB   = bufA + (size_t)N_NODES * D_H1;             // 12.8M floats
  // ACC2 lives directly in d_out (same size), divided in place at the end.

  // 1) zero counts, ACC1, and d_out (self-initializing, graph-replay safe)
  zero_kernel<<<256, 256, 0, stream>>>(counts, 50176);
  zero_kernel<<<2048, 256, 0, stream>>>(bufB, N_NODES * D_H1);
  zero_kernel<<<512, 256, 0, stream>>>(out, N_NODES * D_HEAD);

  // 2) in-degree counts (exact: float adds of 1.0 up to 1.6M < 2^24)
  count_kernel<<<(N_EDGES + 255) / 256, 256, 0, stream>>>(src, counts);

  // 3) GEMM1: T1 = X @ W1cat   (3125 waves, 8 waves/block)
  gemm1_kernel<<<391, 256, 0, stream>>>(X, W1, bufA);

  // 4) edge scatter layer 1: ACC1[src] += T1[dst]
  scatter1_kernel<<<N_EDGES / 8, 256, 0, stream>>>(src, dst, bufA, bufB);

  // 5) H = elu(ACC1 / counts)  (writes over T1 buffer)
  divide_elu_kernel<<<(N_NODES * D_H1 + 255) / 256, 256, 0, stream>>>(bufB, counts, bufA);

  // 6) GEMM2: T2 = H @ W2      (writes over ACC1 buffer)
  gemm2_kernel<<<391, 256, 0, stream>>>(bufA, W2, bufB);

  // 7) edge scatter layer 2: out[src] += T2[dst]
  scatter2_kernel<<<N_EDGES / 8, 256, 0, stream>>>(src, dst, bufB, out);

  // 8) out /= counts
  finalize_kernel<<<(N_NODES * D_HEAD + 255) / 256, 256, 0, stream>>>(out, counts);
}